// IVTLR_45174466019651
// MI455X (gfx1250) — compile-verified
//
#include <hip/hip_runtime.h>
#include <hip/hip_bf16.h>

// Problem constants (match reference)
#define B_     16
#define S_     4096
#define D_     2048
#define NIMG_  3072
#define KMAX_  64
#define NEGF   (-1e30f)
#define ROWS_  4      // output rows pipelined per block in the gather kernel

typedef __attribute__((ext_vector_type(4))) float v4f;   // clang vector: valid for NT builtins

// ---------------------------------------------------------------------------
// CDNA5 async global -> LDS copy (VGLOBAL encoding, tracked by ASYNCcnt).
// vdst = VGPR holding per-lane LDS byte address, vaddr = 64-bit global addr.
// ---------------------------------------------------------------------------
__device__ __forceinline__ void async_load_b128(unsigned lds_off, const void* gptr) {
    asm volatile("global_load_async_to_lds_b128 %0, %1, off"
                 :: "v"(lds_off), "v"(gptr)
                 : "memory");
}
// Wait until at most N async ops remain outstanding (loads complete in order).
__device__ __forceinline__ void wait_asynccnt_le4() {
    asm volatile("s_wait_asynccnt 0x4" ::: "memory");
}
__device__ __forceinline__ void wait_asynccnt_le0() {
    asm volatile("s_wait_asynccnt 0x0" ::: "memory");
}

// Monotone key: higher score wins; on ties, lower position wins (stable argsort(-s)).
__device__ __forceinline__ unsigned long long mk_key(float f, int pos) {
    unsigned u = __float_as_uint(f);
    u = (u & 0x80000000u) ? ~u : (u | 0x80000000u);
    return ((unsigned long long)u << 32) | (unsigned)(0xFFFFFFFFu - (unsigned)pos);
}

// ---------------------------------------------------------------------------
// Kernel 1: per-batch greedy top-k unique-source selection + softmax weights.
// One block per batch, 512 threads (16 wave32s). All candidate state in LDS.
// ---------------------------------------------------------------------------
__global__ __launch_bounds__(512)
void IVTLR_select(const float* __restrict__ scores,
                  const int* __restrict__ source_ids,
                  const unsigned char* __restrict__ init_mask,
                  const unsigned char* __restrict__ new_mask,
                  const int* __restrict__ kptr,
                  int* __restrict__ ws_pos,
                  int* __restrict__ ws_src,
                  float* __restrict__ ws_w)
{
    const int b   = blockIdx.x;
    const int tid = threadIdx.x;
    const int nthr = 512;

    __shared__ float s_score[S_];
    __shared__ int   s_src[S_];
    __shared__ unsigned long long s_red[16];
    __shared__ int   s_selsrc;
    __shared__ int   s_stop;
    __shared__ int   s_selpos[KMAX_];
    __shared__ float s_selscore[KMAX_];

    // Stage masked scores + sources into LDS.
    for (int i = tid; i < S_; i += nthr) {
        const size_t g = (size_t)b * S_ + i;
        float sc = scores[g];
        int  sid = source_ids[g];
        bool m   = (init_mask[g] | new_mask[g]) != 0;
        bool fin = ((__float_as_uint(sc) & 0x7F800000u) != 0x7F800000u);
        bool v   = m && fin && (sid >= 0);
        s_score[i] = v ? sc : NEGF;
        s_src[i]   = sid;
    }
    if (tid < KMAX_) { s_selpos[tid] = -1; s_selscore[tid] = NEGF; }
    if (tid == 0) s_stop = 0;
    __syncthreads();

    int kv = *kptr;
    if (kv > KMAX_) kv = KMAX_;

    for (int j = 0; j < kv; ++j) {
        // Per-thread local best over strided entries.
        unsigned long long best = 0ull;
        for (int i = tid; i < S_; i += nthr) {
            unsigned long long key = mk_key(s_score[i], i);
            if (key > best) best = key;
        }
        // wave32 reduction (warpSize == 32 on gfx1250).
        #pragma unroll
        for (int off = 16; off > 0; off >>= 1) {
            unsigned long long o = __shfl_xor(best, off, 32);
            if (o > best) best = o;
        }
        if ((tid & 31) == 0) s_red[tid >> 5] = best;
        __syncthreads();
        if (tid == 0) {
            unsigned long long bb = s_red[0];
            for (int wv = 1; wv < (nthr >> 5); ++wv)
                if (s_red[wv] > bb) bb = s_red[wv];
            int pos = (int)(0xFFFFFFFFu - (unsigned)(bb & 0xFFFFFFFFull));
            float sc = s_score[pos];
            if (sc > -1e29f) {
                s_selpos[j]   = pos;
                s_selscore[j] = sc;
                s_selsrc      = s_src[pos];
            } else {
                s_selsrc = (int)0x80000000;  // sentinel
                s_stop   = 1;
            }
        }
        __syncthreads();
        if (s_stop) break;
        // Kill every candidate sharing the selected source (dedup semantics).
        const int ss = s_selsrc;
        for (int i = tid; i < S_; i += nthr)
            if (s_src[i] == ss) s_score[i] = NEGF;
        __syncthreads();
    }
    __syncthreads();

    // Softmax over selected slots; invalid slots get weight 0 (exp(-1e30-m)==0).
    if (tid == 0) {
        float m = -3.4e38f;
        for (int j = 0; j < KMAX_; ++j)
            if (s_selpos[j] >= 0 && s_selscore[j] > m) m = s_selscore[j];
        float sum = 0.0f;
        for (int j = 0; j < KMAX_; ++j)
            if (s_selpos[j] >= 0) sum += __expf(s_selscore[j] - m);
        const float inv = (sum > 0.0f) ? (1.0f / sum) : 0.0f;
        const int base = b * KMAX_;
        for (int j = 0; j < KMAX_; ++j) {
            const int p = s_selpos[j];
            float w = 0.0f;
            int sid = 0;
            if (p >= 0) {
                w = __expf(s_selscore[j] - m) * inv;
                sid = s_src[p];
                sid = sid < 0 ? 0 : (sid > (NIMG_ - 1) ? (NIMG_ - 1) : sid);
            }
            ws_pos[base + j] = (p >= 0) ? p : 0;  // safe default for invalid slots
            ws_src[base + j] = sid;
            ws_w[base + j]   = w;
        }
    }
}

// ---------------------------------------------------------------------------
// Kernel 2: bandwidth-bound gather + scale, ASYNCcnt-pipelined.
// Each block produces ROWS_ output rows of D floats. Double-buffered LDS;
// row r+1's async copies are issued before waiting on row r, and
// s_wait_asynccnt 4 releases exactly when row r's 4 copies have landed
// (async loads complete in order). Each lane reads back only its own
// wave's copies -> no block barrier needed. NT stores keep L2 clean.
// ---------------------------------------------------------------------------
__global__ __launch_bounds__(256)
void IVTLR_gather(const float* __restrict__ inputs_embeds,
                  const float* __restrict__ image_embeds,
                  const int* __restrict__ ws_pos,
                  const int* __restrict__ ws_src,
                  const float* __restrict__ ws_w,
                  float* __restrict__ out,
                  int kout)
{
    const int total = B_ * kout;
    const int base  = blockIdx.x * ROWS_;
    const int tid   = threadIdx.x;          // 256 threads, 8 floats each

    __shared__ __align__(16) float sh[2][2 * D_];   // 2 x 16 KiB: [img | inp]

    float        w[ROWS_];
    bool         live[ROWS_];
    const float* gi[ROWS_];
    const float* ge[ROWS_];

    #pragma unroll
    for (int r = 0; r < ROWS_; ++r) {
        int row = base + r;
        live[r] = (row < total);
        int rc  = live[r] ? row : 0;        // clamp to keep addresses safe
        int b   = rc / kout;
        int j   = rc % kout;
        w[r]    = ws_w[b * KMAX_ + j];
        int pos = ws_pos[b * KMAX_ + j];    // 0 for invalid slots
        int src = ws_src[b * KMAX_ + j];    // 0 for invalid slots
        gi[r]   = image_embeds  + ((size_t)b * NIMG_ + src) * D_ + tid * 8;
        ge[r]   = inputs_embeds + ((size_t)b * S_   + pos) * D_ + tid * 8;
    }

    const unsigned la[2] = { (unsigned)(uintptr_t)(const void*)&sh[0][tid * 8],
                             (unsigned)(uintptr_t)(const void*)&sh[1][tid * 8] };
    const unsigned lb[2] = { (unsigned)(uintptr_t)(const void*)&sh[0][D_ + tid * 8],
                             (unsigned)(uintptr_t)(const void*)&sh[1][D_ + tid * 8] };

    // Prologue: stage row 0 into buffer 0.
    async_load_b128(la[0],      gi[0]);
    async_load_b128(la[0] + 16, gi[0] + 4);
    async_load_b128(lb[0],      ge[0]);
    async_load_b128(lb[0] + 16, ge[0] + 4);

    #pragma unroll
    for (int r = 0; r < ROWS_; ++r) {
        const int cur = r & 1;
        if (r + 1 < ROWS_) {
            const int nxt = (r + 1) & 1;
            async_load_b128(la[nxt],      gi[r + 1]);
            async_load_b128(la[nxt] + 16, gi[r + 1] + 4);
            async_load_b128(lb[nxt],      ge[r + 1]);
            async_load_b128(lb[nxt] + 16, ge[r + 1] + 4);
            wait_asynccnt_le4();            // row r's 4 copies have completed
        } else {
            wait_asynccnt_le0();            // drain for the last row
        }

        const v4f* A  = reinterpret_cast<const v4f*>(&sh[cur][tid * 8]);
        const v4f* Bv = reinterpret_cast<const v4f*>(&sh[cur][D_ + tid * 8]);
        v4f a0 = A[0],  a1 = A[1];
        v4f b0 = Bv[0], b1 = Bv[1];

        const float wr = w[r];
        v4f r0, r1;
        if (wr == 0.0f) {                   // invalid slot -> exact zero row
            r0 = (v4f)(0.0f);
            r1 = (v4f)(0.0f);
        } else {
            r0 = wr * (a0 + b0);
            r1 = wr * (a1 + b1);
        }
        if (live[r]) {
            v4f* o4 = reinterpret_cast<v4f*>(out + (size_t)(base + r) * D_) + tid * 2;
            __builtin_nontemporal_store(r0, o4);
            __builtin_nontemporal_store(r1, o4 + 1);
        }
    }
}

// ---------------------------------------------------------------------------
extern "C" void kernel_launch(void* const* d_in, const int* in_sizes, int n_in,
                              void* d_out, int out_size, void* d_ws, size_t ws_size,
                              hipStream_t stream) {
    const float*         scores        = (const float*)d_in[0];
    const float*         inputs_embeds = (const float*)d_in[1];
    const float*         image_embeds  = (const float*)d_in[2];
    const int*           source_ids    = (const int*)d_in[3];
    const unsigned char* init_mask     = (const unsigned char*)d_in[4];
    const unsigned char* new_mask      = (const unsigned char*)d_in[5];
    const int*           kptr          = (const int*)d_in[6];

    int*   ws_pos = (int*)d_ws;
    int*   ws_src = ws_pos + B_ * KMAX_;
    float* ws_w   = (float*)(ws_src + B_ * KMAX_);

    IVTLR_select<<<B_, 512, 0, stream>>>(scores, source_ids, init_mask, new_mask,
                                         kptr, ws_pos, ws_src, ws_w);

    const int kout   = out_size / (B_ * D_);                  // == 64
    const int nblk   = (B_ * kout + ROWS_ - 1) / ROWS_;       // == 256
    IVTLR_gather<<<nblk, 256, 0, stream>>>(inputs_embeds, image_embeds,
                                           ws_pos, ws_src, ws_w,
                                           (float*)d_out, kout);
}